// ForwardSumLoss_15109694947685
// MI455X (gfx1250) — compile-verified
//
#include <hip/hip_runtime.h>
#include <hip/hip_bf16.h>
#include <stdint.h>

// ForwardSumLoss (CTC forward algorithm) for B=64, T=1000, U=250.
// Pass 1: per-(b,t) log-softmax normalizer (wave32 per row, streams 64MB once).
// Pass 2: one 512-thread block per batch chain; async global->LDS 4-slot ring
//         buffering of attn rows (ASYNCcnt) hides memory latency behind the
//         serial lse recursion.
// Pass 3: deterministic tree reduction of per-batch losses.

#define NEGV (-1000000000.0f)
#define BLANKLP (-1.0f)

constexpr int TT = 1000;
constexpr int UU = 250;
constexpr int DEPTH = 4;            // LDS row ring depth (power of two)
constexpr int ROW_INSTRS = 3;       // async instrs per row: 2x b128 + 1x b64 (1000 B)

typedef int v4i_t __attribute__((ext_vector_type(4)));
typedef int v2i_t __attribute__((ext_vector_type(2)));
typedef __attribute__((address_space(1))) v4i_t gv4i;
typedef __attribute__((address_space(3))) v4i_t lv4i;
typedef __attribute__((address_space(1))) v2i_t gv2i;
typedef __attribute__((address_space(3))) v2i_t lv2i;

__device__ __forceinline__ void async_b128(const float* g, float* l) {
#if __has_builtin(__builtin_amdgcn_global_load_async_to_lds_b128)
  __builtin_amdgcn_global_load_async_to_lds_b128(
      (gv4i*)(unsigned long long)g,
      (lv4i*)(unsigned int)(unsigned long long)l, 0, 0);
#else
  *(float4*)l = *(const float4*)g;   // synchronous fallback
#endif
}

__device__ __forceinline__ void async_b64(const float* g, float* l) {
#if __has_builtin(__builtin_amdgcn_global_load_async_to_lds_b64)
  __builtin_amdgcn_global_load_async_to_lds_b64(
      (gv2i*)(unsigned long long)g,
      (lv2i*)(unsigned int)(unsigned long long)l, 0, 0);
#else
  *(float2*)l = *(const float2*)g;   // synchronous fallback
#endif
}

__device__ __forceinline__ void wait_async_hi() {
#if __has_builtin(__builtin_amdgcn_s_wait_asynccnt)
  __builtin_amdgcn_s_wait_asynccnt(ROW_INSTRS * (DEPTH - 1)); // <= 9 outstanding
#else
  asm volatile("s_wait_asynccnt 9" ::: "memory");
#endif
}

__device__ __forceinline__ void wait_async_0() {
#if __has_builtin(__builtin_amdgcn_s_wait_asynccnt)
  __builtin_amdgcn_s_wait_asynccnt(0);
#else
  asm volatile("s_wait_asynccnt 0" ::: "memory");
#endif
}

// Issue one attn row (UU floats = 1000 bytes) global -> LDS slot.
// Executed by one wave (lanes 0..31). Exactly ROW_INSTRS async instructions.
__device__ __forceinline__ void issue_row(const float* __restrict__ g, float* l, int lane) {
  // 62 full float4 chunks: iteration 1 lanes 0..31, iteration 2 lanes 0..29.
  for (int k = lane; k < (UU >> 2); k += 32)
    async_b128(g + 4 * k, l + 4 * k);
  // 2-float tail (floats 248..249)
  if ((UU & 3) && lane == 0)
    async_b64(g + (UU & ~3), l + (UU & ~3));
}

// ---------------- Pass 1: per-row log-softmax normalizer -------------------
// Row r = b*T + t. x[0] = blank (-1.0), x[1..U] = attn[b,0,t,:], columns
// c > text_lens[b] masked out.  norm[r] = max + log(sum exp(x - max)).
__global__ __launch_bounds__(256) void fsl_pass1(const float* __restrict__ attn,
                                                 const int* __restrict__ text_lens,
                                                 float* __restrict__ norm,
                                                 int nrows) {
  int wid = threadIdx.x >> 5;
  int lane = threadIdx.x & 31;
  long long r = (long long)blockIdx.x * 8 + wid;
  if (r >= nrows) return;
  int b = (int)(r / TT);
  int last = text_lens[b];          // valid columns c = 0..last  (last <= U)
  if (last > UU) last = UU;
  const float* row = attn + (size_t)r * UU;

  float m = NEGV;
  for (int c = lane; c <= last; c += 32) {
    float x = (c == 0) ? BLANKLP : row[c - 1];
    m = fmaxf(m, x);
  }
  for (int off = 16; off; off >>= 1) m = fmaxf(m, __shfl_xor(m, off, 32));

  float ssum = 0.0f;
  for (int c = lane; c <= last; c += 32) {
    float x = (c == 0) ? BLANKLP : row[c - 1];
    ssum += __expf(x - m);
  }
  for (int off = 16; off; off >>= 1) ssum += __shfl_xor(ssum, off, 32);

  if (lane == 0) norm[r] = m + __logf(ssum);
}

// ---------------- Pass 2: serial CTC forward recursion ---------------------
// One block per batch element; thread s owns CTC state s (S = 2U+1 = 501).
__global__ __launch_bounds__(512) void fsl_pass2(const float* __restrict__ attn,
                                                 const int* __restrict__ text_lens,
                                                 const int* __restrict__ mel_lens,
                                                 const float* __restrict__ norm,
                                                 float* __restrict__ lossArr) {
  __shared__ float sA[2][514];                 // [0..1] = NEG pads, state s at +2
  __shared__ float sRow[DEPTH][256];           // attn row ring (250 used)
  __shared__ float sNorm[TT];                  // per-t logZ for this batch

  const int b = blockIdx.x;
  const int s = threadIdx.x;
  const int lane = s & 31;
  const int textLen = text_lens[b];
  const int melLen = mel_lens[b];
  const float* rowBase = attn + (size_t)b * TT * UU;
  const float* normB = norm + (size_t)b * TT;

  for (int t = s; t < TT; t += 512) sNorm[t] = normB[t];
  sA[0][s + 2] = (s == 0) ? 0.0f : NEGV;
  if (s < 2) { sA[0][s] = NEGV; sA[1][s] = NEGV; }

  // Pre-fill ring with rows 0..DEPTH-2
  if (s < 32) {
    for (int r0 = 0; r0 < DEPTH - 1; ++r0)
      issue_row(rowBase + (size_t)r0 * UU, &sRow[r0][0], lane);
  }

  const int lab = (s & 1) ? ((s + 1) >> 1) : 0;       // extended label column
  const bool validS = (lab <= textLen);
  const int ci = (lab > 0) ? (lab - 1) : 0;           // safe LDS index

  int cur = 0;
  for (int t = 0; t < melLen; ++t) {
    if (s < 32) {
      int pf = t + DEPTH - 1;
      if (pf < TT) {                                  // slot (t-1)%DEPTH: freed at B2(t-1)
        issue_row(rowBase + (size_t)pf * UU, &sRow[pf & (DEPTH - 1)][0], lane);
        wait_async_hi();                              // rows complete in order -> row t done
      } else {
        wait_async_0();
      }
    }
    __syncthreads();                                  // B1: row t + alpha visible

    const float logZ = sNorm[t];
    const float* row = &sRow[t & (DEPTH - 1)][0];
    float xr = row[ci];
    float xc = (lab == 0) ? BLANKLP : xr;
    float lp = validS ? (xc - logZ) : NEGV;

    float a1 = sA[cur][s + 2];
    float a2 = sA[cur][s + 1];
    float a3 = (s & 1) ? sA[cur][s] : NEGV;
    float m = fmaxf(a1, fmaxf(a2, a3));
    float l = m + __logf(__expf(a1 - m) + __expf(a2 - m) + __expf(a3 - m));
    sA[cur ^ 1][s + 2] = lp + l;

    __syncthreads();                                  // B2: all reads of cur/slot done
    cur ^= 1;
  }

  if (s == 0) {
    int idx = 2 * textLen;
    float aL = sA[cur][idx + 2];
    float aM = sA[cur][idx + 1];
    float m = fmaxf(aL, aM);
    float fin = m + __logf(__expf(aL - m) + __expf(aM - m));
    lossArr[b] = (fin < NEGV * 0.5f) ? 0.0f : (-fin / (float)textLen);
  }
}

// ---------------- Pass 3: deterministic mean -------------------------------
__global__ __launch_bounds__(256) void fsl_pass3(const float* __restrict__ lossArr,
                                                 float* __restrict__ out, int B) {
  __shared__ float sh[256];
  int i = threadIdx.x;
  sh[i] = (i < B) ? lossArr[i] : 0.0f;
  __syncthreads();
  for (int w = 128; w >= 1; w >>= 1) {
    if (i < w) sh[i] += sh[i + w];
    __syncthreads();
  }
  if (i == 0) out[0] = sh[0] / (float)B;
}

extern "C" void kernel_launch(void* const* d_in, const int* in_sizes, int n_in,
                              void* d_out, int out_size, void* d_ws, size_t ws_size,
                              hipStream_t stream) {
  const float* attn = (const float*)d_in[0];
  const int* text_lens = (const int*)d_in[1];
  const int* mel_lens = (const int*)d_in[2];
  float* out = (float*)d_out;

  const int B = in_sizes[1];
  float* norm = (float*)d_ws;                       // B*T floats
  float* lossArr = norm + (size_t)B * TT;           // B floats

  int nrows = B * TT;
  int blocks1 = (nrows + 7) / 8;                    // 8 wave32 rows per 256-thread block
  fsl_pass1<<<blocks1, 256, 0, stream>>>(attn, text_lens, norm, nrows);
  fsl_pass2<<<B, 512, 0, stream>>>(attn, text_lens, mel_lens, norm, lossArr);
  fsl_pass3<<<1, 256, 0, stream>>>(lossArr, out, B);
}